// EEG_GAT_73521250173567
// MI455X (gfx1250) — compile-verified
//
#include <hip/hip_runtime.h>
#include <hip/hip_bf16.h>
#include <math.h>

// f32 WMMA fragment types for V_WMMA_F32_16X16X4_F32 on gfx1250 (wave32):
//   A: 16x4 f32 -> 2 VGPRs (v2f), B: 4x16 f32 -> 2 VGPRs (v2f),
//   C/D: 16x16 f32 -> 8 VGPRs (v8f)
typedef __attribute__((ext_vector_type(2))) float v2f;
typedef __attribute__((ext_vector_type(8))) float v8f;

#define FIN 250
#define FOUT 250
#define NNODES 129024   // 2048 * 63
#define CNODES 63
#define WT_DIM 256      // padded & transposed W: Wt[n][k], 256x256
#define HWS_STRIDE 256  // padded row stride for h workspace

// ---------------------------------------------------------------------------
// Kernel 0: Wt[n][k] = (n<250 && k<250) ? W[k][n] : 0   (zero-pad + transpose)
// Makes every B-fragment load unconditional AND contiguous (b64 per tile).
// ---------------------------------------------------------------------------
__global__ __launch_bounds__(256) void pad_transpose_w_kernel(
    const float* __restrict__ W, float* __restrict__ Wt) {
  const int idx = blockIdx.x * 256 + threadIdx.x;  // 0..65535
  const int n = idx >> 8;
  const int k = idx & 255;
  Wt[idx] = (n < FOUT && k < FIN) ? W[k * FOUT + n] : 0.0f;
}

// ---------------------------------------------------------------------------
// Kernel 1: h = X @ W, out = h + bias. Wave computes 16(M) x 64(N) via 4
// f32 WMMA accumulators. Block = 8 waves -> 128 rows. grid = (1008, 4).
// A frag: lane l: m = l&15, k = k0 + 2*(l>>4) + {0,1}   -> b64 from X row
// B frag: lane l: n = nBase + t*16 + (l&15), k pair     -> b64 from Wt row
// ---------------------------------------------------------------------------
__global__ __launch_bounds__(256) void gat_gemm_kernel(
    const float* __restrict__ X, const float* __restrict__ Wt,
    const float* __restrict__ bias, float* __restrict__ out,
    float* __restrict__ hws) {
  const int lane = threadIdx.x & 31;
  const int wave = threadIdx.x >> 5;           // 0..7
  const int laneHi = lane >> 4;                // 0 or 1
  const int l15 = lane & 15;

  const int rowBase = blockIdx.x * 128 + wave * 16;
  const int nBase = blockIdx.y * 64;

  v8f acc0 = {}, acc1 = {}, acc2 = {}, acc3 = {};

  const float* aP = X + (size_t)(rowBase + l15) * FIN + 2 * laneHi;
  const float* bP = Wt + ((size_t)(nBase + l15) << 8) + 2 * laneHi;

  // 62 unguarded iterations (k0 = 0..244): all loads in-bounds.
#pragma unroll 2
  for (int k0 = 0; k0 < 248; k0 += 4) {
    const v2f a  = *(const v2f*)(aP + k0);
    const v2f b0 = *(const v2f*)(bP + k0);
    const v2f b1 = *(const v2f*)(bP + k0 + 16 * WT_DIM);
    const v2f b2 = *(const v2f*)(bP + k0 + 32 * WT_DIM);
    const v2f b3 = *(const v2f*)(bP + k0 + 48 * WT_DIM);
    acc0 = __builtin_amdgcn_wmma_f32_16x16x4_f32(false, a, false, b0, (short)0, acc0, false, false);
    acc1 = __builtin_amdgcn_wmma_f32_16x16x4_f32(false, a, false, b1, (short)0, acc1, false, false);
    acc2 = __builtin_amdgcn_wmma_f32_16x16x4_f32(false, a, false, b2, (short)0, acc2, false, false);
    acc3 = __builtin_amdgcn_wmma_f32_16x16x4_f32(false, a, false, b3, (short)0, acc3, false, false);
  }

  // Tail k0 = 248: hi lane-half would read k=250,251 (past row end) -> zero A
  // there. B side reads Wt rows 250/251 which are zero-padded already.
  {
    v2f a;
    if (laneHi == 0) {
      a = *(const v2f*)(aP + 248);
    } else {
      a.x = 0.0f; a.y = 0.0f;
    }
    const v2f b0 = *(const v2f*)(bP + 248);
    const v2f b1 = *(const v2f*)(bP + 248 + 16 * WT_DIM);
    const v2f b2 = *(const v2f*)(bP + 248 + 32 * WT_DIM);
    const v2f b3 = *(const v2f*)(bP + 248 + 48 * WT_DIM);
    acc0 = __builtin_amdgcn_wmma_f32_16x16x4_f32(false, a, false, b0, (short)0, acc0, false, false);
    acc1 = __builtin_amdgcn_wmma_f32_16x16x4_f32(false, a, false, b1, (short)0, acc1, false, false);
    acc2 = __builtin_amdgcn_wmma_f32_16x16x4_f32(false, a, false, b2, (short)0, acc2, false, false);
    acc3 = __builtin_amdgcn_wmma_f32_16x16x4_f32(false, a, false, b3, (short)0, acc3, false, false);
  }

  // Epilogue: C layout -> lane holds C[m][n], m = r + 8*laneHi (VGPR r).
  const bool stashH = (rowBase < CNODES);  // uniform per wave (block 0 only)
  auto store_tile = [&](const v8f& acc, int t) {
    const int n = nBase + t * 16 + l15;
    if (n < FOUT) {
      const float bv = bias[n];
#pragma unroll
      for (int r = 0; r < 8; ++r) {
        const int m = rowBase + r + laneHi * 8;
        out[(size_t)m * FOUT + n] = acc[r] + bv;
      }
      if (stashH) {
#pragma unroll
        for (int r = 0; r < 8; ++r) {
          const int m = rowBase + r + laneHi * 8;
          if (m < CNODES) hws[m * HWS_STRIDE + n] = acc[r];  // pre-bias h
        }
      }
    }
  };
  store_tile(acc0, 0);
  store_tile(acc1, 1);
  store_tile(acc2, 2);
  store_tile(acc3, 3);
}

// ---------------------------------------------------------------------------
// Kernel 2: dense 63x63 attention over h[0:63]; overwrites out rows 0..62.
// ---------------------------------------------------------------------------
__global__ __launch_bounds__(256) void gat_attn_kernel(
    const float* __restrict__ hws, const float* __restrict__ att_src,
    const float* __restrict__ att_dst, const float* __restrict__ bias,
    float* __restrict__ out) {
  __shared__ float sA[CNODES];
  __shared__ float sD[CNODES];
  __shared__ float alpha[CNODES][64];

  const int tid = threadIdx.x;

  if (tid < CNODES) {
    float s = 0.0f, d = 0.0f;
    for (int f = 0; f < FOUT; ++f) {
      const float hv = hws[tid * HWS_STRIDE + f];
      s += hv * att_src[f];
      d += hv * att_dst[f];
    }
    sA[tid] = s;
    sD[tid] = d;
  }
  __syncthreads();

  if (tid < CNODES) {
    const float di = sD[tid];
    float m = -INFINITY;
    for (int j = 0; j < CNODES; ++j) {
      float e = sA[j] + di;
      e = (e > 0.0f) ? e : 0.2f * e;   // leaky_relu, slope 0.2
      alpha[tid][j] = e;
      m = fmaxf(m, e);
    }
    float denom = 0.0f;
    for (int j = 0; j < CNODES; ++j) {
      const float ex = __expf(alpha[tid][j] - m);
      alpha[tid][j] = ex;
      denom += ex;
    }
    const float inv = 1.0f / denom;
    for (int j = 0; j < CNODES; ++j) alpha[tid][j] *= inv;
  }
  __syncthreads();

  for (int o = tid; o < CNODES * FOUT; o += 256) {
    const int i = o / FOUT;
    const int f = o - i * FOUT;
    float s = bias[f];
    for (int j = 0; j < CNODES; ++j) s += alpha[i][j] * hws[j * HWS_STRIDE + f];
    out[(size_t)i * FOUT + f] = s;
  }
}

// ---------------------------------------------------------------------------
extern "C" void kernel_launch(void* const* d_in, const int* in_sizes, int n_in,
                              void* d_out, int out_size, void* d_ws,
                              size_t ws_size, hipStream_t stream) {
  (void)in_sizes; (void)n_in; (void)out_size; (void)ws_size;
  const float* x       = (const float*)d_in[0];  // [2048,1,63,250]
  const float* W       = (const float*)d_in[1];  // [250,250]
  const float* att_src = (const float*)d_in[2];  // [250]
  const float* att_dst = (const float*)d_in[3];  // [250]
  const float* bias    = (const float*)d_in[4];  // [250]
  float* out = (float*)d_out;                    // [129024,250]

  float* Wt  = (float*)d_ws;                     // 256*256*4 = 256 KB
  float* hws = (float*)d_ws + WT_DIM * WT_DIM;   // 63*256*4  = 64.5 KB

  pad_transpose_w_kernel<<<WT_DIM * WT_DIM / 256, 256, 0, stream>>>(W, Wt);
  dim3 grid(NNODES / 128, 4);                    // (1008, 4)
  gat_gemm_kernel<<<grid, 256, 0, stream>>>(x, Wt, bias, out, hws);
  gat_attn_kernel<<<1, 256, 0, stream>>>(hws, att_src, att_dst, bias, out);
}